// PointPillarsScatter_15719580304030
// MI455X (gfx1250) — compile-verified
//
#include <hip/hip_runtime.h>
#include <hip/hip_bf16.h>
#include <stdint.h>

// PointPillarsScatter: dense [N,C,H,W] canvas from M pillar features.
#define NN  4
#define CC  64
#define HH  496
#define WW  432
#define HW  (HH * WW)         // 214272
#define CHW (CC * HW)         // 13713408
#define TOT (NN * CHW)        // 54853632 floats = 219.4 MB
#define MM  40000

// ---------------------------------------------------------------------------
// Kernel 1: zero the canvas via the CDNA5 async LDS->global store path.
// Each lane keeps 16 B of zeros in LDS and streams them out with
// global_store_async_from_lds_b128 (ASYNCcnt), drained by s_wait_asynccnt.
// 13392 blocks * 256 threads * 4 stores * 16 B = 219.4 MB exactly (no tail).
// ---------------------------------------------------------------------------
__global__ void __launch_bounds__(256) pps_zero_async(float* __restrict__ out) {
    __shared__ float4 zbuf[256];
    zbuf[threadIdx.x] = make_float4(0.f, 0.f, 0.f, 0.f);
    __syncthreads();

    // Low 32 bits of the flat address of an LDS object are its LDS byte offset.
    uint32_t lds_off = (uint32_t)(uintptr_t)(&zbuf[threadIdx.x]);

    const uint32_t T = gridDim.x * 256u;               // total threads
    const uint32_t t = blockIdx.x * 256u + threadIdx.x;
    float4* o = (float4*)out;

#pragma unroll
    for (uint32_t k = 0; k < 4; ++k) {
        float4* gp = o + (t + k * T);                  // lanes -> adjacent 16B
        asm volatile("global_store_async_from_lds_b128 %0, %1, off"
                     :: "v"(gp), "v"(lds_off)
                     : "memory");
    }
    // Drain async stores before LDS is deallocated at wave end.
    asm volatile("s_wait_asynccnt 0" ::: "memory");
}

// ---------------------------------------------------------------------------
// Kernel 2: scatter. 16 lanes per pillar; each lane loads a float4 of 4
// consecutive channels (fully coalesced B128 reads from [M,64]) and writes
// 4 floats at stride H*W into the canvas. Coords are unique -> no conflicts.
// global_prefetch_b8 warms the streaming feature reads.
// ---------------------------------------------------------------------------
__global__ void __launch_bounds__(256) pps_scatter(const float* __restrict__ feat,
                                                   const int*   __restrict__ coords,
                                                   float*       __restrict__ out) {
    const uint32_t tid = blockIdx.x * 256u + threadIdx.x;   // < 640000 exactly
    const uint32_t m  = tid >> 4;                            // pillar id
    const uint32_t c0 = (tid & 15u) << 2;                    // first channel

    // coords row: (n, z, y, x), 16-byte aligned int4
    const int4 cd = *(const int4*)(coords + 4u * m);

    // Prefetch features ~256 pillars ahead (clamped to stay in-buffer).
    uint32_t pm = m + 256u;
    if (pm >= MM) pm = m;
    __builtin_prefetch(feat + (size_t)pm * CC, 0, 1);

    const float4 v = *(const float4*)(feat + (size_t)m * CC + c0);

    const uint32_t base = (uint32_t)cd.x * (uint32_t)CHW
                        + c0 * (uint32_t)HW
                        + (uint32_t)cd.z * (uint32_t)WW
                        + (uint32_t)cd.w;

    out[base           ] = v.x;
    out[base +     HW  ] = v.y;
    out[base + 2u * HW ] = v.z;
    out[base + 3u * HW ] = v.w;
}

// ---------------------------------------------------------------------------
extern "C" void kernel_launch(void* const* d_in, const int* in_sizes, int n_in,
                              void* d_out, int out_size, void* d_ws, size_t ws_size,
                              hipStream_t stream) {
    const float* feat   = (const float*)d_in[0];   // [M, C] f32
    const int*   coords = (const int*)  d_in[1];   // [M, 4] i32
    float*       out    = (float*)d_out;           // [N, C, H, W] f32

    // 13392 * 256 threads * 4 float4 = 13,713,408 float4 = whole canvas.
    pps_zero_async<<<13392, 256, 0, stream>>>(out);

    // 2500 * 256 = 640,000 = M * 16 lanes.
    pps_scatter<<<2500, 256, 0, stream>>>(feat, coords, out);
}